// InferenceAttention_1760936591726
// MI455X (gfx1250) — compile-verified
//
#include <hip/hip_runtime.h>

// ---------------- problem constants (match reference) ----------------
#define B_     16
#define N_     16
#define D_     4096
#define H_     32
#define HK_    8
#define DH_    128
#define CTX_   4096
#define START_ 2048
#define T_     (START_ + N_)        // 2064
#define REP_   (H_ / HK_)           // 4
#define TS_    4                    // attention time-split chunks
#define TILES_T ((T_ + 15) / 16)    // 129
#define TILES_PER_CHUNK ((TILES_T + TS_ - 1) / TS_)  // 33
#define SCALE_ 0.08838834764831845f // 1/sqrt(128)

typedef float v2f __attribute__((ext_vector_type(2)));
typedef float v8f __attribute__((ext_vector_type(8)));
typedef unsigned int u32x4 __attribute__((ext_vector_type(4)));
typedef int i32x4v __attribute__((ext_vector_type(4)));
typedef int i32x8v __attribute__((ext_vector_type(8)));

__device__ __forceinline__ v8f wmma_f32(v2f a, v2f b, v8f c) {
  // D = A(16x4) * B(4x16) + C(16x16), fp32, wave32
  return __builtin_amdgcn_wmma_f32_16x16x4_f32(false, a, false, b, (short)0, c,
                                               false, false);
}

// TDM: load a 16-row x 128-col f32 tile (row stride `rowStride` elements)
// from global `src` into LDS byte offset `ldsOff`. Wave-level op; EXEC
// ignored; tracked by TENSORcnt.
__device__ __forceinline__ void tdm_load_tile_16x128(const float* src,
                                                     unsigned ldsOff,
                                                     unsigned rowStride) {
  unsigned long long ga = (unsigned long long)(uintptr_t)src;
  u32x4 g0;
  g0[0] = 1u;                                   // count=1, user descriptor
  g0[1] = ldsOff;                               // lds_addr
  g0[2] = (unsigned)ga;                         // global_addr[31:0]
  g0[3] = (unsigned)((ga >> 32) & 0x01FFFFFFu)  // global_addr[56:32]
          | 0x80000000u;                        // type=2 ("image")
  i32x8v g1;
  g1[0] = 0x20000;        // data_size=2 (4 bytes); no mask/pad/iterate
  g1[1] = (int)(128u << 16);   // tensor_dim0 = 128 (bits 79:48 low half)
  g1[2] = (int)(16u << 16);    // tensor_dim1 = 16
  g1[3] = (int)(128u << 16);   // tile_dim0 = 128
  g1[4] = 16;                  // tile_dim1 = 16, tile_dim2 = 0
  g1[5] = (int)rowStride;      // tensor_dim0_stride (elements)
  g1[6] = 0;
  g1[7] = 0;
  i32x4v z4 = {0, 0, 0, 0};
#if __clang_major__ >= 23
  i32x8v z8 = {0, 0, 0, 0, 0, 0, 0, 0};
  __builtin_amdgcn_tensor_load_to_lds(g0, g1, z4, z4, z8, 0);
#else
  __builtin_amdgcn_tensor_load_to_lds(g0, g1, z4, z4, 0);
#endif
}

// =====================================================================
// Generic fp32 WMMA GEMM: C[M x Nout] = A[M x K] * W[K x Nout]
// blockDim = (32, 8). Each wave computes a 16x32 strip (two 16x16 tiles)
// stepping K by 8 -> four independent WMMA accumulation chains; A
// fragments shared between the two N-tiles.
// =====================================================================
__global__ __launch_bounds__(256) void gemm_f32_wmma(
    const float* __restrict__ A, const float* __restrict__ W,
    float* __restrict__ C, int K, int Nout) {
  const int lane = threadIdx.x;            // 0..31
  const int hb   = lane >> 4;              // lane half
  const int lr   = lane & 15;
  const int kb   = hb * 2;                 // K sub-pair within a 4-step

  const int col0 = (blockIdx.x * 8 + threadIdx.y) * 32;
  const int row0 = blockIdx.y * 16;

  const float* aPtr = A + (size_t)(row0 + lr) * K;   // A row per lane
  const float* w0   = W + col0 + lr;                 // N-tile 0 column
  const float* w1   = w0 + 16;                       // N-tile 1 column

  v8f acc00 = {}, acc01 = {}, acc10 = {}, acc11 = {};
  for (int k = 0; k < K; k += 8) {
    v2f a0, a1;
    a0.x = aPtr[k + kb];
    a0.y = aPtr[k + kb + 1];
    a1.x = aPtr[k + 4 + kb];
    a1.y = aPtr[k + 4 + kb + 1];
    const size_t r0 = (size_t)(k + kb) * Nout;
    const size_t r1 = r0 + Nout;
    const size_t r2 = (size_t)(k + 4 + kb) * Nout;
    const size_t r3 = r2 + Nout;
    v2f b00, b10, b01, b11;
    b00.x = w0[r0]; b00.y = w0[r1];
    b10.x = w1[r0]; b10.y = w1[r1];
    b01.x = w0[r2]; b01.y = w0[r3];
    b11.x = w1[r2]; b11.y = w1[r3];
    acc00 = wmma_f32(a0, b00, acc00);   // 4 independent chains
    acc10 = wmma_f32(a0, b10, acc10);
    acc01 = wmma_f32(a1, b01, acc01);
    acc11 = wmma_f32(a1, b11, acc11);
  }
  const v8f accA = acc00 + acc01;
  const v8f accB = acc10 + acc11;

  // C/D layout: VGPR r -> row (row0 + r + 8*half), col (col0 + lr)
  float* cPtr = C + (size_t)(row0 + hb * 8) * Nout + col0 + lr;
#pragma unroll
  for (int r = 0; r < 8; ++r) {
    cPtr[(size_t)r * Nout]      = accA[r];
    cPtr[(size_t)r * Nout + 16] = accB[r];
  }
}

// =====================================================================
// RoPE: t is (B, N, heads, DH); pairs (2j, 2j+1) rotated by angle[n][j].
// =====================================================================
__global__ __launch_bounds__(256) void rope_f32(
    float* __restrict__ t, const float* __restrict__ fcos,
    const float* __restrict__ fsin, int heads) {
  const int half = DH_ / 2;
  int idx = blockIdx.x * blockDim.x + threadIdx.x;
  int total = B_ * N_ * heads * half;
  if (idx >= total) return;
  int j    = idx % half;
  int rest = idx / half;
  int n    = (rest / heads) % N_;
  float c = fcos[n * half + j];
  float s = fsin[n * half + j];
  float* p = t + (size_t)idx * 2;   // == rest*DH + 2*j
  float a = p[0], b = p[1];
  p[0] = a * c - b * s;
  p[1] = a * s + b * c;
}

// =====================================================================
// Flash attention chunk kernel.
// grid = (B, HK, TS_); block = (32, 4): wave w handles query head kh*4+w.
// K/V 16x128 tiles staged in LDS by the Tensor Data Mover (wave 0 issues
// the descriptors; TENSORcnt + barrier for visibility). START is
// 16-aligned so a tile never straddles cache / new-token sources.
// Dynamic LDS layout (bytes): [0,8K) K tile, [8K,16K) V tile,
// [16K + wave*1K) P-transpose scratch.
// =====================================================================
extern __shared__ float smem[];

__global__ __launch_bounds__(128) void attn_flash_chunk(
    const float* __restrict__ xq, const float* __restrict__ xk,
    const float* __restrict__ xv, const float* __restrict__ cache_k,
    const float* __restrict__ cache_v, float* __restrict__ zpart,
    float* __restrict__ mpart, float* __restrict__ lpart) {
  const int lane = threadIdx.x;
  const int wave = threadIdx.y;          // rep index 0..3
  const int b    = blockIdx.x;
  const int kh   = blockIdx.y;
  const int chnk = blockIdx.z;
  const int h    = kh * REP_ + wave;

  const int hb = lane >> 4, lr = lane & 15, kb = hb * 2;

  float* lds_k = smem;                        // 2048 floats
  float* lds_v = smem + 2048;                 // 2048 floats
  float* lds_p = smem + 4096 + wave * 256;    // 256 floats per wave

  // Preload q A-fragments for all 32 K-steps (16 rows x 128 dh).
  v2f qa[32];
  {
    const float* qrow = xq + ((size_t)(b * N_ + lr) * H_ + h) * DH_;
#pragma unroll
    for (int s = 0; s < 32; ++s) {
      int dh = s * 4 + kb;
      qa[s].x = qrow[dh];
      qa[s].y = qrow[dh + 1];
    }
  }

  v8f accz[8];
  float rmax[8], rsum[8];
#pragma unroll
  for (int d = 0; d < 8; ++d) accz[d] = (v8f){};
#pragma unroll
  for (int r = 0; r < 8; ++r) { rmax[r] = -3.0e38f; rsum[r] = 0.0f; }

  const int t0 = chnk * TILES_PER_CHUNK;
  const int t1 = min(TILES_T, t0 + TILES_PER_CHUNK);

  for (int tt = t0; tt < t1; ++tt) {
    const int tbase = tt * 16;

    // ---- TDM: stage K and V 16x128 tiles into LDS ----
    if (wave == 0) {
      const float *ksrc, *vsrc;
      unsigned stride;
      if (tbase < START_) {
        size_t off = (((size_t)b * HK_ + kh) * CTX_ + tbase) * DH_;
        ksrc = cache_k + off;
        vsrc = cache_v + off;
        stride = DH_;              // cache row stride (elements)
      } else {
        size_t off = ((size_t)(b * N_ + (tbase - START_)) * HK_ + kh) * DH_;
        ksrc = xk + off;
        vsrc = xv + off;
        stride = HK_ * DH_;        // (B,N,HK,DH) token stride (elements)
      }
      tdm_load_tile_16x128(ksrc, 0u, stride);
      tdm_load_tile_16x128(vsrc, 8192u, stride);
      __builtin_amdgcn_s_wait_tensorcnt(0);
    }
    __syncthreads();

    // ---- S(16x16) = q * k^T : B[kk][t] = K[t][kk] from LDS ----
    v8f s = {};
    {
      const float* kl = lds_k + lr * DH_;   // column t = tbase + lr
#pragma unroll
      for (int ks = 0; ks < 32; ++ks) {
        v2f bf;
        bf.x = kl[ks * 4 + kb];
        bf.y = kl[ks * 4 + kb + 1];
        s = wmma_f32(qa[ks], bf, s);
      }
    }

    // ---- scale + causal mask + online softmax ----
    const int tcol = tbase + lr;
    float corr[8];
#pragma unroll
    for (int r = 0; r < 8; ++r) {
      const int mrow = r + hb * 8;        // query index 0..15
      float sv = s[r] * SCALE_;
      sv += (tcol <= START_ + mrow) ? 0.0f : -1000000000.0f;  // == mask
      float mx = sv;
      mx = fmaxf(mx, __shfl_xor(mx, 1, 16));
      mx = fmaxf(mx, __shfl_xor(mx, 2, 16));
      mx = fmaxf(mx, __shfl_xor(mx, 4, 16));
      mx = fmaxf(mx, __shfl_xor(mx, 8, 16));
      float mnew = fmaxf(rmax[r], mx);
      corr[r] = __expf(rmax[r] - mnew);
      rmax[r] = mnew;
      float p = __expf(sv - mnew);
      s[r] = p;
      float sm = p;
      sm += __shfl_xor(sm, 1, 16);
      sm += __shfl_xor(sm, 2, 16);
      sm += __shfl_xor(sm, 4, 16);
      sm += __shfl_xor(sm, 8, 16);
      rsum[r] = rsum[r] * corr[r] + sm;
    }
#pragma unroll
    for (int d = 0; d < 8; ++d)
#pragma unroll
      for (int r = 0; r < 8; ++r) accz[d][r] *= corr[r];

    // ---- transpose P through per-wave LDS into A-fragments ----
    {
#pragma unroll
      for (int r = 0; r < 8; ++r) lds_p[(r + hb * 8) * 16 + lr] = s[r];
      // same-wave LDS ops are in-order: safe to read back
      v2f pa[4];
#pragma unroll
      for (int ks = 0; ks < 4; ++ks) {
        pa[ks].x = lds_p[lr * 16 + ks * 4 + kb];
        pa[ks].y = lds_p[lr * 16 + ks * 4 + kb + 1];
      }
      // ---- z(16x128) += P(16x16) * V(16x128), 8 dh-tiles ----
#pragma unroll
      for (int d = 0; d < 8; ++d) {
        const float* vl = lds_v + d * 16 + lr;   // col dh = d*16+lr
#pragma unroll
        for (int ks = 0; ks < 4; ++ks) {
          v2f bf;
          bf.x = vl[(ks * 4 + kb) * DH_];
          bf.y = vl[(ks * 4 + kb + 1) * DH_];
          accz[d] = wmma_f32(pa[ks], bf, accz[d]);
        }
      }
    }
    __syncthreads();   // protect lds_k/lds_v before next tile overwrite
  }

  // ---- emit chunk partials ----
  const int bh   = b * H_ + h;
  const int base = (bh * TS_ + chnk) * 16;
  if (lr == 0) {
#pragma unroll
    for (int r = 0; r < 8; ++r) {
      mpart[base + r + hb * 8] = rmax[r];
      lpart[base + r + hb * 8] = rsum[r];
    }
  }
#pragma unroll
  for (int d = 0; d < 8; ++d)
#pragma unroll
    for (int r = 0; r < 8; ++r) {
      int mrow = r + hb * 8;
      zpart[(size_t)(base + mrow) * DH_ + d * 16 + lr] = accz[d][r];
    }
}

// =====================================================================
// Flash combine: merge TS_ chunk partials, normalize, write z in
// (B, N, H, DH) layout (== (B,N,D), input of the output projection).
// =====================================================================
__global__ __launch_bounds__(128) void attn_combine(
    const float* __restrict__ zpart, const float* __restrict__ mpart,
    const float* __restrict__ lpart, float* __restrict__ z) {
  const int bh   = blockIdx.x >> 4;
  const int mrow = blockIdx.x & 15;
  const int dh   = threadIdx.x;

  float m = -3.0e38f;
#pragma unroll
  for (int c = 0; c < TS_; ++c)
    m = fmaxf(m, mpart[(bh * TS_ + c) * 16 + mrow]);

  float lsum = 0.0f, acc = 0.0f;
#pragma unroll
  for (int c = 0; c < TS_; ++c) {
    int idx = (bh * TS_ + c) * 16 + mrow;
    float w = __expf(mpart[idx] - m);
    lsum += lpart[idx] * w;
    acc  += zpart[(size_t)idx * DH_ + dh] * w;
  }
  const int b = bh / H_, h = bh % H_;
  z[((size_t)(b * N_ + mrow) * H_ + h) * DH_ + dh] = acc / lsum;
}

// =====================================================================
extern "C" void kernel_launch(void* const* d_in, const int* in_sizes, int n_in,
                              void* d_out, int out_size, void* d_ws,
                              size_t ws_size, hipStream_t stream) {
  const float* x       = (const float*)d_in[0];
  const float* fcos    = (const float*)d_in[1];
  const float* fsin    = (const float*)d_in[2];
  // d_in[3] = mask: reproduced arithmetically inside the kernel
  const float* cache_k = (const float*)d_in[4];
  const float* cache_v = (const float*)d_in[5];
  const float* wq      = (const float*)d_in[6];
  const float* wk      = (const float*)d_in[7];
  const float* wv      = (const float*)d_in[8];
  const float* wo      = (const float*)d_in[9];
  float* out = (float*)d_out;

  // workspace layout (floats)
  float* ws = (float*)d_ws;
  float* xq = ws;                                   // 256*4096
  float* xk = xq + (size_t)256 * 4096;              // 256*1024
  float* xv = xk + (size_t)256 * 1024;              // 256*1024
  float* z  = xv + (size_t)256 * 1024;              // 256*4096
  float* zp = z  + (size_t)256 * 4096;              // 512*TS_*16*128
  float* mp = zp + (size_t)B_ * H_ * TS_ * 16 * DH_;
  float* lp = mp + (size_t)B_ * H_ * TS_ * 16;

  const dim3 gblk(32, 8);
  // QKV projections (each wave does a 16x32 strip => grid.x = Nout/256)
  gemm_f32_wmma<<<dim3(D_ / 256, (B_ * N_) / 16), gblk, 0, stream>>>(
      x, wq, xq, D_, H_ * DH_);
  gemm_f32_wmma<<<dim3((HK_ * DH_) / 256, (B_ * N_) / 16), gblk, 0, stream>>>(
      x, wk, xk, D_, HK_ * DH_);
  gemm_f32_wmma<<<dim3((HK_ * DH_) / 256, (B_ * N_) / 16), gblk, 0, stream>>>(
      x, wv, xv, D_, HK_ * DH_);
  // RoPE
  rope_f32<<<(B_ * N_ * H_ * (DH_ / 2)) / 256, 256, 0, stream>>>(xq, fcos,
                                                                 fsin, H_);
  rope_f32<<<(B_ * N_ * HK_ * (DH_ / 2)) / 256, 256, 0, stream>>>(xk, fcos,
                                                                  fsin, HK_);
  // Flash attention (chunked over time, TDM-staged tiles) + combine
  attn_flash_chunk<<<dim3(B_, HK_, TS_), dim3(32, 4), 20480, stream>>>(
      xq, xk, xv, cache_k, cache_v, zp, mp, lp);
  attn_combine<<<B_ * H_ * 16, 128, 0, stream>>>(zp, mp, lp, z);
  // Output projection
  gemm_f32_wmma<<<dim3(D_ / 256, (B_ * N_) / 16), gblk, 0, stream>>>(
      z, wo, out, D_, D_);
}